// BitAllocationNetwork_84748294685452
// MI455X (gfx1250) — compile-verified
//
#include <hip/hip_runtime.h>
#include <math.h>

#define BSZ   4096
#define DD    4096
#define GG    8
#define GSZ   512
#define TWO_G 16

typedef float v2f __attribute__((ext_vector_type(2)));
typedef float v4f __attribute__((ext_vector_type(4)));
typedef float v8f __attribute__((ext_vector_type(8)));

// ---------------------------------------------------------------------------
// Kernel 1: group means. One block per row (256 thr = 8 wave32); each wave
// handles one 512-col group with coalesced 16B non-temporal loads (data is
// read exactly once -> don't pollute L2), then a wave32 shuffle reduction.
// 64 MB streamed read -> HBM bound.
// ---------------------------------------------------------------------------
__global__ void group_mean_kernel(const float* __restrict__ imp,
                                  float* __restrict__ gi)
{
    const int r    = blockIdx.x;
    const int g    = threadIdx.x >> 5;   // group 0..7 (one wave per group)
    const int lane = threadIdx.x & 31;

    const v4f* p = (const v4f*)(imp + (size_t)r * DD + g * GSZ);
    float s = 0.0f;
#pragma unroll
    for (int i = 0; i < 4; ++i) {        // 4 iters * 32 lanes * 4 floats = 512
        v4f v = __builtin_nontemporal_load(&p[i * 32 + lane]);
        s += v.x + v.y + v.z + v.w;
    }
#pragma unroll
    for (int off = 16; off > 0; off >>= 1)
        s += __shfl_down(s, off, 32);
    if (lane == 0)
        gi[(size_t)r * GG + g] = s * (1.0f / (float)GSZ);
}

// ---------------------------------------------------------------------------
// Kernel 2: per-row MLP + alloc + budget discretize. One thread per row.
// All weights are tiny (<1KB) and stay cache-resident.
// ---------------------------------------------------------------------------
__device__ __forceinline__ int level_idx(float b) {
    // b is exactly 2, 4 or 8
    return (b < 3.0f) ? 0 : (b < 6.0f ? 1 : 2);
}

__global__ void mlp_discretize_kernel(const float* __restrict__ gi,
                                      const float* __restrict__ W1,
                                      const float* __restrict__ b1,
                                      const float* __restrict__ gamma,
                                      const float* __restrict__ beta,
                                      const float* __restrict__ W2,
                                      const float* __restrict__ b2,
                                      float* __restrict__ alloc_out,
                                      float* __restrict__ bits_out)
{
    const int r = blockIdx.x * blockDim.x + threadIdx.x;
    if (r >= BSZ) return;

    const float L0 = 2.0f, L1 = 4.0f, L2 = 8.0f;
    const float LV[3] = {L0, L1, L2};

    float g[GG];
#pragma unroll
    for (int i = 0; i < GG; ++i) g[i] = gi[(size_t)r * GG + i];

    // h = gelu(g @ W1 + b1), exact erf gelu
    float h[TWO_G];
    float mu = 0.0f;
#pragma unroll
    for (int j = 0; j < TWO_G; ++j) {
        float acc = b1[j];
#pragma unroll
        for (int i = 0; i < GG; ++i) acc = fmaf(g[i], W1[i * TWO_G + j], acc);
        float hv = 0.5f * acc * (1.0f + erff(acc * 0.70710678118654752f));
        h[j] = hv;
        mu += hv;
    }
    mu *= (1.0f / (float)TWO_G);
    float var = 0.0f;
#pragma unroll
    for (int j = 0; j < TWO_G; ++j) { float d = h[j] - mu; var += d * d; }
    var *= (1.0f / (float)TWO_G);
    const float inv = rsqrtf(var + 1e-5f);
#pragma unroll
    for (int j = 0; j < TWO_G; ++j)
        h[j] = (h[j] - mu) * inv * gamma[j] + beta[j];

    // logits = h @ W2 + b2 ; softmax
    float lg[GG];
    float mx = -1e30f;
#pragma unroll
    for (int k = 0; k < GG; ++k) {
        float acc = b2[k];
#pragma unroll
        for (int j = 0; j < TWO_G; ++j) acc = fmaf(h[j], W2[j * GG + k], acc);
        lg[k] = acc;
        mx = fmaxf(mx, acc);
    }
    float es = 0.0f;
#pragma unroll
    for (int k = 0; k < GG; ++k) { lg[k] = expf(lg[k] - mx); es += lg[k]; }
    const float einv = 1.0f / es;

    // alloc = clip( (2 + p*6) * 32/sum , 2, 8 )
    float a[GG];
    float asum = 0.0f;
#pragma unroll
    for (int k = 0; k < GG; ++k) { a[k] = 2.0f + (lg[k] * einv) * 6.0f; asum += a[k]; }
    const float sc = (4.0f * (float)GG) / asum;
#pragma unroll
    for (int k = 0; k < GG; ++k) {
        a[k] = fminf(8.0f, fmaxf(2.0f, a[k] * sc));
        alloc_out[(size_t)r * GG + k] = a[k];
    }

    // --- discretize with budget ---
    // init: nearest level, first-on-ties
    float bb[GG];
#pragma unroll
    for (int k = 0; k < GG; ++k) {
        float x = a[k];
        float best = fabsf(x - L0); int bi = 0;
        float d1 = fabsf(x - L1); if (d1 < best) { best = d1; bi = 1; }
        float d2 = fabsf(x - L2); if (d2 < best) { best = d2; bi = 2; }
        bb[k] = LV[bi];
    }

    const float BIG = 1e30f;
    for (int it = 0; it < GG * 3; ++it) {
        float s = 0.0f;
        for (int k = 0; k < GG; ++k) s += bb[k];
        const float budget = s * (1.0f / (float)GG);
        if (fabsf(budget - 4.0f) <= 0.1f) break;

        if (budget > 4.0f) {
            // reduce branch
            const float diff = budget - 4.0f;
            float best = BIG; int ri = 0; bool any = false;
            float rv[GG];
            for (int k = 0; k < GG; ++k) {
                int lvl = level_idx(bb[k]);
                bool can = lvl > 0;
                float red = LV[(lvl > 0) ? (lvl - 1) : 0];
                rv[k] = red;
                float imp = (bb[k] - red) * (1.0f / (float)GG);
                float score = can ? fabsf(imp - diff) : BIG;
                if (score < best) { best = score; ri = k; }
                any |= can;
            }
            if (any) bb[ri] = rv[ri];
        } else {
            // increase branch
            const float diff = 4.0f - budget;
            float best = BIG; int ii = 0; bool any = false;
            float iv[GG];
            for (int k = 0; k < GG; ++k) {
                int lvl = level_idx(bb[k]);
                bool can = lvl < 2;
                float inc = LV[(lvl < 2) ? (lvl + 1) : 2];
                iv[k] = inc;
                float imp = (inc - bb[k]) * (1.0f / (float)GG);
                float score = can ? fabsf(imp - diff) : BIG;
                if (score < best) { best = score; ii = k; }
                any |= can;
            }
            if (any) bb[ii] = iv[ii];
        }
    }

#pragma unroll
    for (int k = 0; k < GG; ++k) bits_out[(size_t)r * GG + k] = bb[k];
}

// ---------------------------------------------------------------------------
// Kernel 3: bit_embeddings = softmax(exp(-|bits-L|)) @ codebook, via
// V_WMMA_F32_16X16X4_F32.  M = 32768 bg rows, K = 3 (padded to 4), N = 512.
// One block = 16 bg-rows; 4 waves each cover 128 of the 512 columns.
// A layout (16x4 f32): lanes 0-15 hold {K0,K1}, lanes 16-31 hold {K2,K3}.
// C layout (16x16 f32): VGPR v, lane l -> (M = v + 8*(l>=16), N = l%16).
//
// K=3 pad handling: A's K3 slot (amat.y on lanes 16-31) is exactly 0, so B's
// K3 slot only needs to be FINITE (0*finite == 0; 0*Inf/NaN would poison).
// We therefore load codebook row 2 (finite) into it unconditionally, which
// keeps every load uniform across the wave -> no EXEC-mask branches in the
// hot loop (the previous version emitted s_and_saveexec/cbranch per tile).
//
// Output is written once and never re-read on device -> non-temporal stores.
// 64 MB streamed store -> HBM bound; WMMA does the K-reduction for free.
// ---------------------------------------------------------------------------
__global__ void embed_wmma_kernel(const float* __restrict__ bits,
                                  const float* __restrict__ cb,
                                  float* __restrict__ emb)
{
    const int tile = blockIdx.x;            // 16 bg rows per tile
    const int lane = threadIdx.x & 31;
    const int wave = threadIdx.x >> 5;      // 0..3 -> column quarter
    const int half = lane >> 4;             // 0: K0/K1 lanes, 1: K2/K3 lanes
    const int m    = lane & 15;

    const int bg0 = tile * 16;

    // per-row mixture weights: softmax over exp(-|bits - L|)
    const float bv = bits[bg0 + m];
    const float t0 = expf(-fabsf(bv - 2.0f));
    const float t1 = expf(-fabsf(bv - 4.0f));
    const float t2 = expf(-fabsf(bv - 8.0f));
    const float tm = fmaxf(t0, fmaxf(t1, t2));
    const float e0 = expf(t0 - tm);
    const float e1 = expf(t1 - tm);
    const float e2 = expf(t2 - tm);
    const float winv = 1.0f / (e0 + e1 + e2);
    const float w0 = e0 * winv, w1 = e1 * winv, w2 = e2 * winv;

    v2f amat;
    amat.x = half ? w2 : w0;     // K2 : K0
    amat.y = half ? 0.0f : w1;   // K3(pad, exact 0) : K1

    // uniform (branch-free) codebook row selectors
    const int row_x = half ? 2 : 0;   // K2 : K0
    const int row_y = half ? 2 : 1;   // K3 dummy (finite, multiplied by 0) : K1

    float* __restrict__ out = emb + (size_t)bg0 * GSZ;

#pragma unroll
    for (int ct = 0; ct < 8; ++ct) {
        const int col = wave * 128 + ct * 16 + m;

        v2f bmat;
        bmat.x = cb[row_x * GSZ + col];
        bmat.y = cb[row_y * GSZ + col];

        v8f c = {};
        c = __builtin_amdgcn_wmma_f32_16x16x4_f32(
                /*neg_a=*/false, amat,
                /*neg_b=*/false, bmat,
                /*c_mod=*/(short)0, c,
                /*reuse_a=*/false, /*reuse_b=*/false);

#pragma unroll
        for (int v = 0; v < 8; ++v) {
            const int row = v + 8 * half;
            __builtin_nontemporal_store(c[v], &out[(size_t)row * GSZ + col]);
        }
    }
}

// ---------------------------------------------------------------------------
extern "C" void kernel_launch(void* const* d_in, const int* in_sizes, int n_in,
                              void* d_out, int out_size, void* d_ws, size_t ws_size,
                              hipStream_t stream)
{
    const float* imp   = (const float*)d_in[0];   // [4096,4096]
    const float* W1    = (const float*)d_in[1];   // [8,16]
    const float* b1    = (const float*)d_in[2];   // [16]
    const float* gamma = (const float*)d_in[3];   // [16]
    const float* beta  = (const float*)d_in[4];   // [16]
    const float* W2    = (const float*)d_in[5];   // [16,8]
    const float* b2    = (const float*)d_in[6];   // [8]
    const float* cb    = (const float*)d_in[7];   // [3,512]

    float* out       = (float*)d_out;
    float* alloc_out = out;                       // [4096,8]
    float* bits_out  = out + BSZ * GG;            // [4096,8]
    float* emb_out   = out + 2 * BSZ * GG;        // [4096,8,512]

    // stash group means at the start of the embeddings region; kernel 2 reads
    // them, kernel 3 fully overwrites the region (stream-ordered, so safe and
    // deterministic -- no dependence on ws_size).
    float* gi = emb_out;

    hipLaunchKernelGGL(group_mean_kernel, dim3(BSZ), dim3(256), 0, stream,
                       imp, gi);
    hipLaunchKernelGGL(mlp_discretize_kernel, dim3(BSZ / 256), dim3(256), 0, stream,
                       gi, W1, b1, gamma, beta, W2, b2, alloc_out, bits_out);
    hipLaunchKernelGGL(embed_wmma_kernel, dim3((BSZ * GG) / 16), dim3(128), 0, stream,
                       bits_out, cb, emb_out);
}